// DynamicDWConv_64656437674380
// MI455X (gfx1250) — compile-verified
//
#include <hip/hip_runtime.h>
#include <stdint.h>

#define B_    8
#define DIM_  256
#define SEQ_  16384
#define K_    3
#define HID_  85
#define DIMK_ (DIM_ * K_)   // 768
#define EPS_  1e-5f
#define CTILE 1024

typedef __attribute__((ext_vector_type(2))) float v2f;
typedef __attribute__((ext_vector_type(8))) float v8f;

// ---------------- Kernel 1: mean over seq -> pooled[B, DIM] ----------------
__global__ __launch_bounds__(256) void pool_kernel(const float* __restrict__ x,
                                                   float* __restrict__ pooled) {
    const int row = blockIdx.x;                       // b*DIM + c
    const float4* xr = (const float4*)(x + (size_t)row * SEQ_);
    float s = 0.f;
    for (int i = threadIdx.x; i < SEQ_ / 4; i += 256) {
        float4 v = xr[i];
        s += (v.x + v.y) + (v.z + v.w);
    }
    __shared__ float red[256];
    red[threadIdx.x] = s;
    __syncthreads();
    for (int off = 128; off > 0; off >>= 1) {
        if (threadIdx.x < off) red[threadIdx.x] += red[threadIdx.x + off];
        __syncthreads();
    }
    if (threadIdx.x == 0) pooled[row] = red[0] * (1.0f / SEQ_);
}

// ------- Kernel 2: WMMA GEMM1 + BN + ReLU + WMMA GEMM2 -> wk[B, 768] -------
// One block, 256 threads (8 waves). Wave 0: y = ReLU(BN(pooled @ w1^T)) into LDS.
// All waves: wk = y @ w2^T + b2 (48 N-tiles of 16, 6 per wave).
// Padding strategy: clamp row/index and load unconditionally (no exec-masked
// loads); every padded output element is overwritten with an exact 0, and the
// K-padding of GEMM2 is zero on the A side, so clamped-garbage B values are
// multiplied by exact zeros.
__global__ __launch_bounds__(256) void gen_kernel(
    const float* __restrict__ pooled,  // [8,256]
    const float* __restrict__ gamma_, const float* __restrict__ beta_,
    const float* __restrict__ mean_,  const float* __restrict__ var_,
    const float* __restrict__ w1,      // [85,256]
    const float* __restrict__ w2,      // [768,85]
    const float* __restrict__ b2,      // [768]
    float* __restrict__ wk)            // [8,768]
{
    __shared__ float yls[16][96];      // y padded: rows 8..15 = 0, cols 85..95 = 0

    const int lane = threadIdx.x & 31;
    const int wave = threadIdx.x >> 5;
    const int half = (lane < 16) ? 0 : 2;   // K sub-offset for this half-wave
    const int mn   = lane & 15;             // M index (A) / N index (B,C,D)

    if (wave == 0) {
        // GEMM1: [16 x 256] * [256 x 96] -> [16 x 96]
        const float* ar = pooled + (mn & 7) * DIM_;          // rows 8..15 alias 0..7
        for (int nt = 0; nt < 6; ++nt) {
            const int n  = nt * 16 + mn;
            const int nc = (n < HID_) ? n : (HID_ - 1);      // clamped column
            const float* br = w1 + nc * DIM_;
            v8f c = {};
            for (int kb = 0; kb < 256; kb += 4) {
                v2f a = *(const v2f*)(ar + kb + half);       // 8B-aligned b64
                v2f b = *(const v2f*)(br + kb + half);       // 8B-aligned b64
                c = __builtin_amdgcn_wmma_f32_16x16x4_f32(
                        false, a, false, b, (short)0, c, false, false);
            }
            // BN + ReLU (clamped params; padded entries -> literal 0)
            const float sc = gamma_[nc] * rsqrtf(var_[nc] + EPS_);
            const float mu = mean_[nc];
            const float bt = beta_[nc];
            const bool live = (lane < 16) && (n < HID_);
            for (int v = 0; v < 8; ++v) {
                const int m = v + ((lane < 16) ? 0 : 8);
                const float yv = live ? fmaxf(fmaf(c[v] - mu, sc, bt), 0.f) : 0.f;
                yls[m][nt * 16 + mn] = yv;
            }
        }
    }
    __syncthreads();

    // GEMM2: [16 x 96] * [96 x 768] -> [16 x 768]; y cols 85..95 are exact 0
    for (int tt = 0; tt < 6; ++tt) {
        const int nt = wave * 6 + tt;
        const int n  = nt * 16 + mn;            // always < 768
        const float* w2r = w2 + n * HID_;
        v8f c = {};
        for (int kb = 0; kb < 96; kb += 4) {
            const int k0 = kb + half;
            const int k1 = k0 + 1;
            v2f a = *(const v2f*)&yls[mn][k0];   // 8B-aligned ds b64
            v2f b;
            b.x = w2r[(k0 < HID_) ? k0 : (HID_ - 1)];   // unconditional b32
            b.y = w2r[(k1 < HID_) ? k1 : (HID_ - 1)];   // unconditional b32
            c = __builtin_amdgcn_wmma_f32_16x16x4_f32(
                    false, a, false, b, (short)0, c, false, false);
        }
        if (lane < 16) {
            const float bb = b2[n];
            for (int v = 0; v < 8; ++v) {        // rows m = 0..7 live in lanes 0..15
                wk[v * DIMK_ + n] = c[v] + bb;
            }
        }
    }
}

// ------- Kernel 3: dynamic depthwise conv (K=3, pad=1) + bias, streaming -------
// Tile staged into LDS via GLOBAL_LOAD_ASYNC_TO_LDS_B128 (ASYNCcnt path).
__global__ __launch_bounds__(256) void conv_kernel(const float* __restrict__ x,
                                                   const float* __restrict__ wk,
                                                   const float* __restrict__ bias,
                                                   float* __restrict__ out) {
    const int row = blockIdx.y;            // b*DIM + c
    const int cch = row & (DIM_ - 1);
    const int bb  = row >> 8;              // DIM_ = 256
    const size_t base = (size_t)row * SEQ_;
    const int s0 = blockIdx.x * CTILE;

    // tile[4 + j] = x[s0 + j] for j in [0, CTILE); halo: tile[3] = x[s0-1],
    // tile[4 + CTILE] = x[s0 + CTILE]. tile[4] is 16B-aligned.
    __shared__ __align__(16) float tile[CTILE + 8];

    const int t = threadIdx.x;
    // LDS aperture: generic-pointer low 32 bits are the LDS byte offset.
    const unsigned ldst = ((unsigned)(uintptr_t)tile) + 16u + 16u * (unsigned)t;
    const uint64_t gsrc = (uint64_t)(uintptr_t)(x + base + s0 + 4 * (size_t)t);
    // 16B global -> 16B LDS per lane, tracked by ASYNCcnt.
    asm volatile("global_load_async_to_lds_b128 %0, %1, off"
                 :: "v"(ldst), "v"(gsrc) : "memory");

    if (t == 0) tile[3]         = (s0 > 0)            ? x[base + s0 - 1]     : 0.f;
    if (t == 1) tile[4 + CTILE] = (s0 + CTILE < SEQ_) ? x[base + s0 + CTILE] : 0.f;

    const float w0 = wk[bb * DIMK_ + cch * K_ + 0];
    const float w1 = wk[bb * DIMK_ + cch * K_ + 1];
    const float w2 = wk[bb * DIMK_ + cch * K_ + 2];
    const float bs = bias[cch];

    asm volatile("s_wait_asynccnt 0x0" ::: "memory");
    __syncthreads();

    const int o = t * 4;                   // 4 outputs per thread
    float4 r;
    r.x = fmaf(w0, tile[o + 3], fmaf(w1, tile[o + 4], fmaf(w2, tile[o + 5], bs)));
    r.y = fmaf(w0, tile[o + 4], fmaf(w1, tile[o + 5], fmaf(w2, tile[o + 6], bs)));
    r.z = fmaf(w0, tile[o + 5], fmaf(w1, tile[o + 6], fmaf(w2, tile[o + 7], bs)));
    r.w = fmaf(w0, tile[o + 6], fmaf(w1, tile[o + 7], fmaf(w2, tile[o + 8], bs)));
    *(float4*)(out + base + s0 + o) = r;
}

extern "C" void kernel_launch(void* const* d_in, const int* in_sizes, int n_in,
                              void* d_out, int out_size, void* d_ws, size_t ws_size,
                              hipStream_t stream) {
    const float* x     = (const float*)d_in[0];
    const float* w1    = (const float*)d_in[1];
    const float* gamma = (const float*)d_in[2];
    const float* beta  = (const float*)d_in[3];
    const float* mean  = (const float*)d_in[4];
    const float* var   = (const float*)d_in[5];
    const float* w2    = (const float*)d_in[6];
    const float* b2    = (const float*)d_in[7];
    const float* bias  = (const float*)d_in[8];
    float* out = (float*)d_out;

    float* pooled = (float*)d_ws;                  // [8,256]  = 8 KB
    float* wkern  = pooled + B_ * DIM_;            // [8,768]  = 24 KB

    pool_kernel<<<B_ * DIM_, 256, 0, stream>>>(x, pooled);
    gen_kernel<<<1, 256, 0, stream>>>(pooled, gamma, beta, mean, var, w1, w2, b2, wkern);
    conv_kernel<<<dim3(SEQ_ / CTILE, B_ * DIM_), 256, 0, stream>>>(x, wkern, bias, out);
}